// YOLOv8Loss_4767413698929
// MI455X (gfx1250) — compile-verified
//
#include <hip/hip_runtime.h>
#include <math.h>

#define TPB   256
#define NANCH 8400
#define NCLS  80
#define NBINS 16
#define NB    16
#define NG    32
#define EPSF  1e-9f

typedef __attribute__((ext_vector_type(2))) float v2f;
typedef __attribute__((ext_vector_type(8))) float v8f;

// ---------------- wave32 sum reduction: matrix-pipe version -----------------
// A (16x4 f32): lanes 0-15 hold rows M=0..15 at K0(K1), lanes 16-31 at K2(K3).
// With a = {v, 0} and B = ones, D[m][j] = v[m] + v[m+16] (pairsum).
// C/D layout: VGPR r, lanes 0-15 -> M=r; lanes 16-31 -> M=r+8.
__device__ inline float wave_sum(float v) {
#if __has_builtin(__builtin_amdgcn_wmma_f32_16x16x4_f32)
    v2f a; a.x = v;    a.y = 0.0f;
    v2f b; b.x = 1.0f; b.y = 1.0f;
    v8f c = {};
    v8f d = __builtin_amdgcn_wmma_f32_16x16x4_f32(
        false, a, false, b, (short)0, c, false, false);
    float s = d[0] + d[1] + d[2] + d[3] + d[4] + d[5] + d[6] + d[7];
    // lanes<16 hold sum(pairsum[0..7]), lanes>=16 hold sum(pairsum[8..15])
    s += __shfl_xor(s, 16, 32);
    return s;
#else
    for (int o = 16; o > 0; o >>= 1) v += __shfl_xor(v, o, 32);
    return v;
#endif
}

// ---------------- geometry / math helpers -----------------------------------
struct Anc { const float* base; int hw; int loc; float cx, cy, st; };

__device__ inline Anc anc_info(const float* p0, const float* p1, const float* p2,
                               int b, int n) {
    Anc a;
    if (n < 6400) {
        a.base = p0 + (size_t)b * 144 * 6400; a.hw = 6400; a.loc = n;
        int W = 80; a.st = 8.0f;
        int x = a.loc % W, y = a.loc / W;
        a.cx = (x + 0.5f) * a.st; a.cy = (y + 0.5f) * a.st;
    } else if (n < 8000) {
        int l = n - 6400;
        a.base = p1 + (size_t)b * 144 * 1600; a.hw = 1600; a.loc = l;
        int W = 40; a.st = 16.0f;
        int x = l % W, y = l / W;
        a.cx = (x + 0.5f) * a.st; a.cy = (y + 0.5f) * a.st;
    } else {
        int l = n - 8000;
        a.base = p2 + (size_t)b * 144 * 400; a.hw = 400; a.loc = l;
        int W = 20; a.st = 32.0f;
        int x = l % W, y = l / W;
        a.cx = (x + 0.5f) * a.st; a.cy = (y + 0.5f) * a.st;
    }
    return a;
}

__device__ inline float sigmoidf(float x) { return 1.0f / (1.0f + expf(-x)); }
__device__ inline float pow6f(float x) { float x2 = x * x; return x2 * x2 * x2; }

__device__ inline float pair_iou(const float* g, const float* p) {
    float ltx = fmaxf(g[0], p[0]), lty = fmaxf(g[1], p[1]);
    float rbx = fminf(g[2], p[2]), rby = fminf(g[3], p[3]);
    float w = fmaxf(rbx - ltx, 0.0f), h = fmaxf(rby - lty, 0.0f);
    float inter = w * h;
    float ag = (g[2] - g[0]) * (g[3] - g[1]);
    float ap = fmaxf(p[2] - p[0], 0.0f) * fmaxf(p[3] - p[1], 0.0f);
    return inter / (ag + ap - inter + EPSF);
}

__device__ inline bool in_gt(const Anc& a, const float* g) {
    return (a.cx > g[0]) && (a.cx < g[2]) && (a.cy > g[1]) && (a.cy < g[3]);
}

__device__ inline float ciou_loss(const float* p, const float* t) {
    float w1 = p[2] - p[0], h1 = p[3] - p[1];
    float w2 = t[2] - t[0], h2 = t[3] - t[1];
    float iw = fmaxf(fminf(p[2], t[2]) - fmaxf(p[0], t[0]), 0.0f);
    float ih = fmaxf(fminf(p[3], t[3]) - fmaxf(p[1], t[1]), 0.0f);
    float inter = iw * ih;
    float uni = w1 * h1 + w2 * h2 - inter + EPSF;
    float iou = inter / uni;
    float cw = fmaxf(p[2], t[2]) - fminf(p[0], t[0]);
    float chh = fmaxf(p[3], t[3]) - fminf(p[1], t[1]);
    float c2 = cw * cw + chh * chh + EPSF;
    float dx = p[0] + p[2] - t[0] - t[2];
    float dy = p[1] + p[3] - t[1] - t[3];
    float rho2 = (dx * dx + dy * dy) * 0.25f;
    float da = atanf(w2 / (h2 + EPSF)) - atanf(w1 / (h1 + EPSF));
    const float KPI = 4.0f / (float)(M_PI * M_PI);
    float v = KPI * da * da;
    float alpha = v / (v - iou + 1.0f + EPSF);
    return 1.0f - (iou - rho2 / c2 - v * alpha);
}

// ---------------- kernel 0: init accumulators --------------------------------
__global__ void k_init(float* accum, float* amax, float* omax) {
    int i = threadIdx.x;
    if (i < 4) accum[i] = 0.0f;
    for (int j = i; j < NB * NG; j += TPB) { amax[j] = 0.0f; omax[j] = 0.0f; }
}

// ---------------- kernel 1: DFL softmax -> decoded boxes ---------------------
__global__ void k_pred(const float* p0, const float* p1, const float* p2,
                       float* predb) {
    int idx = blockIdx.x * TPB + threadIdx.x;
    if (idx >= NB * NANCH) return;
    int b = idx / NANCH, n = idx % NANCH;
    Anc a = anc_info(p0, p1, p2, b, n);
    float dist[4];
#pragma unroll
    for (int s = 0; s < 4; ++s) {
        float l[NBINS]; float m = -1e30f;
#pragma unroll
        for (int k = 0; k < NBINS; ++k) {
            l[k] = a.base[(size_t)(s * NBINS + k) * a.hw + a.loc];
            m = fmaxf(m, l[k]);
        }
        float se = 0.0f, sw = 0.0f;
#pragma unroll
        for (int k = 0; k < NBINS; ++k) {
            float e = expf(l[k] - m);
            se += e; sw += e * (float)k;
        }
        dist[s] = (sw / se) * a.st;
    }
    float* o = predb + (size_t)idx * 4;
    o[0] = a.cx - dist[0]; o[1] = a.cy - dist[1];
    o[2] = a.cx + dist[2]; o[3] = a.cy + dist[3];
}

// ---------------- kernel 2: per-(b,g) TAL metric top-10 threshold ------------
__global__ void k_topk(const float* p0, const float* p1, const float* p2,
                       const float* predb, const float* gtb, const int* gtl,
                       float* kth) {
    int bg = blockIdx.x;
    int b = bg / NG;
    int tid = threadIdx.x;
    float g[4];
    g[0] = gtb[(size_t)bg * 4 + 0]; g[1] = gtb[(size_t)bg * 4 + 1];
    g[2] = gtb[(size_t)bg * 4 + 2]; g[3] = gtb[(size_t)bg * 4 + 3];
    int lab = gtl[bg];

    float t10[10];
#pragma unroll
    for (int j = 0; j < 10; ++j) t10[j] = -1.0f;

    for (int n = tid; n < NANCH; n += TPB) {
        Anc a = anc_info(p0, p1, p2, b, n);
        const float* pb = predb + ((size_t)b * NANCH + n) * 4;
        float metric = 0.0f;
        if (in_gt(a, g)) {
            float iou = pair_iou(g, pb);
            float sc = sigmoidf(a.base[(size_t)(64 + lab) * a.hw + a.loc]);
            metric = sqrtf(sc) * pow6f(iou);
        }
        if (metric > t10[9]) {
            int j = 9;
            while (j > 0 && t10[j - 1] < metric) { t10[j] = t10[j - 1]; --j; }
            t10[j] = metric;
        }
    }

    __shared__ float cand[TPB * 10];
    __shared__ float rv[TPB];
    __shared__ int   ri[TPB];
#pragma unroll
    for (int j = 0; j < 10; ++j) cand[tid * 10 + j] = t10[j];
    __syncthreads();

    float kthv = 0.0f;
    for (int r = 0; r < 10; ++r) {
        float lv = -2.0f; int li = -1;
#pragma unroll
        for (int j = 0; j < 10; ++j) {
            float v = cand[tid * 10 + j];
            if (v > lv) { lv = v; li = tid * 10 + j; }
        }
        rv[tid] = lv; ri[tid] = li;
        __syncthreads();
        for (int s = TPB / 2; s > 0; s >>= 1) {
            if (tid < s && rv[tid + s] > rv[tid]) {
                rv[tid] = rv[tid + s]; ri[tid] = ri[tid + s];
            }
            __syncthreads();
        }
        if (tid == 0) { kthv = rv[0]; if (ri[0] >= 0) cand[ri[0]] = -2.0f; }
        __syncthreads();
    }
    if (tid == 0) kth[bg] = kthv;
}

// ---------------- kernel 3: assignment (mask, multi-fix, matched/fg) ---------
__global__ void k_assign(const float* p0, const float* p1, const float* p2,
                         const float* predb, const float* gtb, const int* gtl,
                         const float* kth, int* matched, int* fgm,
                         float* amax, float* omax, float* accum) {
    int b = blockIdx.y;
    int n = blockIdx.x * TPB + threadIdx.x;
    __shared__ float sg[NG][4];
    __shared__ int   sl[NG];
    __shared__ float sk[NG];
    if (threadIdx.x < NG) {
        int g = threadIdx.x;
        for (int i = 0; i < 4; ++i) sg[g][i] = gtb[((size_t)b * NG + g) * 4 + i];
        sl[g] = gtl[b * NG + g];
        sk[g] = kth[b * NG + g];
    }
    __syncthreads();

    bool f = false;
    if (n < NANCH) {
        Anc a = anc_info(p0, p1, p2, b, n);
        const float* pbp = predb + ((size_t)b * NANCH + n) * 4;
        float pb[4] = { pbp[0], pbp[1], pbp[2], pbp[3] };

        int cnt = 0, firstg = -1, bestg = 0;
        float bestiou = -1.0f;
        for (int g = 0; g < NG; ++g) {
            float iou = pair_iou(sg[g], pb);
            float met = 0.0f;
            if (in_gt(a, sg[g])) {
                float sc = sigmoidf(a.base[(size_t)(64 + sl[g]) * a.hw + a.loc]);
                met = sqrtf(sc) * pow6f(iou);
            }
            bool mp = (met >= sk[g]) && (met > EPSF);
            if (mp) { ++cnt; if (firstg < 0) firstg = g; }
            if (iou > bestiou) { bestiou = iou; bestg = g; }
        }
        int m;
        if (cnt > 1) { m = bestg; f = true; }
        else         { f = (cnt == 1); m = f ? firstg : 0; }
        matched[(size_t)b * NANCH + n] = m;
        fgm[(size_t)b * NANCH + n] = f ? 1 : 0;

        if (f) {
            float iou_m = pair_iou(sg[m], pb);
            float met_m = 0.0f;
            if (in_gt(a, sg[m])) {
                float sc = sigmoidf(a.base[(size_t)(64 + sl[m]) * a.hw + a.loc]);
                met_m = sqrtf(sc) * pow6f(iou_m);
            }
            atomicMax((unsigned int*)&amax[b * NG + m], __float_as_uint(met_m));
            atomicMax((unsigned int*)&omax[b * NG + m], __float_as_uint(iou_m));
        }
    }
    // fg count via matrix-unit wave reduction
    float c = wave_sum(f ? 1.0f : 0.0f);
    if ((threadIdx.x & 31) == 0) atomicAdd(&accum[3], c);
}

// ---------------- kernel 5: losses (CIoU, DFL, focal) ------------------------
__global__ void k_loss(const float* p0, const float* p1, const float* p2,
                       const float* predb, const float* gtb, const int* gtl,
                       const int* matched, const int* fgm,
                       const float* amax, const float* omax, float* accum) {
    int b = blockIdx.y;
    int n = blockIdx.x * TPB + threadIdx.x;
    __shared__ float sg[NG][4];
    __shared__ int   sl[NG];
    if (threadIdx.x < NG) {
        int g = threadIdx.x;
        for (int i = 0; i < 4; ++i) sg[g][i] = gtb[((size_t)b * NG + g) * 4 + i];
        sl[g] = gtl[b * NG + g];
    }
    __syncthreads();

    float liou = 0.0f, ldfl = 0.0f, lcls = 0.0f;
    if (n < NANCH) {
        Anc a = anc_info(p0, p1, p2, b, n);
        const float* pbp = predb + ((size_t)b * NANCH + n) * 4;
        float pb[4] = { pbp[0], pbp[1], pbp[2], pbp[3] };
        int  m = matched[(size_t)b * NANCH + n];
        bool f = fgm[(size_t)b * NANCH + n] != 0;
        float tb[4] = { sg[m][0], sg[m][1], sg[m][2], sg[m][3] };
        int   tl = sl[m];

        // norm factor (ts magnitude)
        float nrm = 0.0f;
        if (f) {
            float iou = pair_iou(tb, pb);
            float met = 0.0f;
            if (in_gt(a, tb)) {
                float sc = sigmoidf(a.base[(size_t)(64 + tl) * a.hw + a.loc]);
                met = sqrtf(sc) * pow6f(iou);
            }
            nrm = met * omax[b * NG + m] / (amax[b * NG + m] + EPSF);
        }

        // CIoU
        if (f) liou = ciou_loss(pb, tb);

        // DFL: target dists + log-softmax cross-entropy
        {
            float dd[4] = { fmaxf(a.cx - tb[0], 0.0f), fmaxf(a.cy - tb[1], 0.0f),
                            fmaxf(tb[2] - a.cx, 0.0f), fmaxf(tb[3] - a.cy, 0.0f) };
            float contrib = 0.0f;
#pragma unroll
            for (int s = 0; s < 4; ++s) {
                float d = fminf(fmaxf(dd[s] / a.st, 0.0f),
                                (float)(NBINS - 1) - 1e-6f);
                float lo = floorf(d);
                float al = d - lo;
                int lo_i = (int)lo;
                int up_i = min(lo_i + 1, NBINS - 1);
                float l[NBINS]; float mx = -1e30f;
#pragma unroll
                for (int k = 0; k < NBINS; ++k) {
                    l[k] = a.base[(size_t)(s * NBINS + k) * a.hw + a.loc];
                    mx = fmaxf(mx, l[k]);
                }
                float se = 0.0f;
#pragma unroll
                for (int k = 0; k < NBINS; ++k) se += expf(l[k] - mx);
                float lse = mx + logf(se);
                contrib -= (1.0f - al) * (l[lo_i] - lse) + al * (l[up_i] - lse);
            }
            ldfl = f ? contrib : 0.0f;
        }

        // focal classification over 80 classes
        for (int c = 0; c < NCLS; ++c) {
            float lg = a.base[(size_t)(64 + c) * a.hw + a.loc];
            float t = (f && c == tl) ? nrm : 0.0f;
            float p = sigmoidf(lg);
            float ce = fmaxf(lg, 0.0f) - lg * t + log1pf(expf(-fabsf(lg)));
            float pt = p * t + (1.0f - p) * (1.0f - t);
            float at = t * 0.25f + (1.0f - t) * 0.75f;
            float om = 1.0f - pt;
            lcls += at * om * om * ce;
        }
    }

    float wi = wave_sum(liou);
    float wd = wave_sum(ldfl);
    float wc = wave_sum(lcls);
    if ((threadIdx.x & 31) == 0) {
        atomicAdd(&accum[0], wi);
        atomicAdd(&accum[1], wd);
        atomicAdd(&accum[2], wc);
    }
}

// ---------------- kernel 6: finalize -----------------------------------------
__global__ void k_final(const float* accum, float* out) {
    if (threadIdx.x == 0 && blockIdx.x == 0) {
        float den = fmaxf(accum[3], 1.0f);
        out[0] = 7.5f * accum[0] / den;   // W_IOU * loss_iou
        out[1] = 1.5f * accum[1] / den;   // W_DFL * loss_dfl
        out[2] = 0.5f * accum[2] / den;   // W_CLS * loss_cls
    }
}

// ---------------- host launch ------------------------------------------------
extern "C" void kernel_launch(void* const* d_in, const int* in_sizes, int n_in,
                              void* d_out, int out_size, void* d_ws, size_t ws_size,
                              hipStream_t stream) {
    const float* p0  = (const float*)d_in[0];
    const float* p1  = (const float*)d_in[1];
    const float* p2  = (const float*)d_in[2];
    const float* gtb = (const float*)d_in[3];
    const int*   gtl = (const int*)d_in[4];
    float* out = (float*)d_out;

    float* ws      = (float*)d_ws;
    float* accum   = ws;                            // 4
    float* predb   = accum + 4;                     // NB*NANCH*4 = 537600
    float* kth     = predb + (size_t)NB * NANCH * 4;// 512
    float* amax    = kth + NB * NG;                 // 512
    float* omax    = amax + NB * NG;                // 512
    int*   matched = (int*)(omax + NB * NG);        // 134400
    int*   fgm     = matched + NB * NANCH;          // 134400

    k_init<<<1, TPB, 0, stream>>>(accum, amax, omax);

    int tot = NB * NANCH;
    k_pred<<<(tot + TPB - 1) / TPB, TPB, 0, stream>>>(p0, p1, p2, predb);

    k_topk<<<NB * NG, TPB, 0, stream>>>(p0, p1, p2, predb, gtb, gtl, kth);

    dim3 gbn((NANCH + TPB - 1) / TPB, NB);
    k_assign<<<gbn, TPB, 0, stream>>>(p0, p1, p2, predb, gtb, gtl, kth,
                                      matched, fgm, amax, omax, accum);

    k_loss<<<gbn, TPB, 0, stream>>>(p0, p1, p2, predb, gtb, gtl,
                                    matched, fgm, amax, omax, accum);

    k_final<<<1, 1, 0, stream>>>(accum, out);
}